// skipP_25512105738439
// MI455X (gfx1250) — compile-verified
//
#include <hip/hip_runtime.h>
#include <math.h>
#include <stdint.h>

// Problem constants (from reference setup_inputs)
namespace {
constexpr int B  = 8;
constexpr int C  = 64;
constexpr int H  = 256;
constexpr int W  = 256;
constexpr int CR = 128;   // readout channels
constexpr int HH = 128;   // readout H (half res)
constexpr int WH = 128;   // readout W (half res)
}

// align_corners 2x upsample coefficients, bit-matching the JAX reference:
//   pos = p * f32((N-1)/(2N-1)); i0 = clip(floor(pos), 0, N-2); w = pos - i0
__device__ __forceinline__ void lerp_coef(int p, int size_in, int& i0, float& w) {
    const float scale = (float)(size_in - 1) / (float)(2 * size_in - 1); // f32(127/255)
    float pos = (float)p * scale;
    int i = (int)pos;                 // pos >= 0, so trunc == floor
    if (i > size_in - 2) i = size_in - 2;
    i0 = i;
    w  = pos - (float)i;
}

// ---- gfx1250 async global->LDS staging -------------------------------------
typedef __attribute__((address_space(1))) int g_as_int;
typedef __attribute__((address_space(3))) int l_as_int;

__device__ __forceinline__ void stage_b32(const float* g, float* lds) {
#if __has_builtin(__builtin_amdgcn_global_load_async_to_lds_b32)
    __builtin_amdgcn_global_load_async_to_lds_b32(
        (g_as_int*)(void*)g,      // global source (address_space(1))
        (l_as_int*)(void*)lds,    // LDS destination (address_space(3))
        0, 0);
#else
    *lds = *g;   // fallback: plain load + ds_store
#endif
}

__device__ __forceinline__ void stage_wait() {
#if __has_builtin(__builtin_amdgcn_global_load_async_to_lds_b32)
#if __has_builtin(__builtin_amdgcn_s_wait_asynccnt)
    __builtin_amdgcn_s_wait_asynccnt(0);
#else
    asm volatile("s_wait_asynccnt 0x0" ::: "memory");
#endif
#endif
}

// sigmoid gate + blend: x*(1-g) + r*g, g = sigmoid(x*w0 + r*w1 + b)
__device__ __forceinline__ float gate_blend(float xv, float rv,
                                            float w0, float w1, float bb) {
    const float z = xv * w0 + rv * w1 + bb;
    const float g = 1.0f / (1.0f + __expf(-z));   // expf(+inf)=inf -> g=0: safe
    return xv * (1.0f - g) + rv * g;
}

// ---- Kernel 1: column sums of the 1D upsample matrix, scaled by 1/(2N) -----
// c[i] = (sum over output positions p of the weight p contributes to input i) / 2N
// Then mean(upsample2d(A)) == sum_ij c[i]*c[j]*A[i,j]   (separable, rank-1)
__global__ void __launch_bounds__(HH) k_colsum(float* __restrict__ cw) {
    int i = threadIdx.x;  // 0..127
    float acc = 0.0f;
    for (int p = 0; p < 2 * HH; ++p) {
        int i0; float w; lerp_coef(p, HH, i0, w);
        if (i0 == i)     acc += (1.0f - w);
        if (i0 + 1 == i) acc += w;
    }
    cw[i] = acc * (1.0f / (2.0f * HH));
}

// ---- Kernel 2: scores[b,cr] = c^T * A * c over each 128x128 slice ----------
__global__ void __launch_bounds__(256) k_scores(const float* __restrict__ readout,
                                                const float* __restrict__ cw,
                                                float* __restrict__ scores) {
    __shared__ float c[WH];
    __shared__ float red[256];
    const int bc = blockIdx.x;          // 0 .. B*CR-1
    const int t  = threadIdx.x;
    if (t < WH) c[t] = cw[t];
    __syncthreads();

    const float4* base4 = (const float4*)(readout + (size_t)bc * (HH * WH));
    float acc = 0.0f;
    for (int q = t; q < (HH * WH) / 4; q += 256) {   // 16 iterations
        const int e   = q * 4;
        const int row = e >> 7;
        const int col = e & (WH - 1);                // col..col+3 same row
        const float4 v = base4[q];
        acc += c[row] * (c[col] * v.x + c[col + 1] * v.y +
                         c[col + 2] * v.z + c[col + 3] * v.w);
    }
    red[t] = acc;
    __syncthreads();
    for (int s = 128; s > 0; s >>= 1) {
        if (t < s) red[t] += red[t + s];
        __syncthreads();
    }
    if (t == 0) scores[bc] = red[0];
}

// ---- Kernel 3: ordered top-64 by exact rank counting -----------------------
// Matches jax.lax.top_k tie-breaking: equal values -> lower index first.
__global__ void __launch_bounds__(CR) k_topk(const float* __restrict__ scores,
                                             int* __restrict__ idx) {
    __shared__ float s[CR];
    const int b = blockIdx.x;
    const int t = threadIdx.x;          // channel 0..127
    s[t] = scores[b * CR + t];
    __syncthreads();
    const float mine = s[t];
    int rank = 0;
    for (int o = 0; o < CR; ++o) {
        float so = s[o];
        rank += (so > mine) || (so == mine && o < t);
    }
    if (rank < C) idx[b * C + rank] = t;
}

// ---- Kernel 4: fused upsample + gate + blend -------------------------------
// One block per 4-row output tile of one (b,c): 256 threads, thread t handles
// output row h0 + (t>>6), pixels w = 4*(t&63) .. +3 (float4 load/store).
// Source rows ih(h0) .. min(ih(h0)+3, 127) cover every per-row stencil
// {ih(h), ih(h)+1} for h in [h0, h0+3] and are contiguous in memory.
__global__ void __launch_bounds__(256) k_fused(const float* __restrict__ x,
                                               const float* __restrict__ readout,
                                               const float* __restrict__ weight,
                                               const float* __restrict__ bias,
                                               const int* __restrict__ idx,
                                               float* __restrict__ out) {
    __shared__ float srow[4 * WH];      // up to 4 staged source rows (2 KB)

    const int t   = threadIdx.x;
    const int bid = blockIdx.x;         // = ((b*C + c)*(H/4) + hq)
    const int hq = bid & (H / 4 - 1);
    const int c  = (bid >> 6) & (C - 1);
    const int b  = bid >> 12;
    const int h0 = hq * 4;

    const int sel = idx[b * C + c];     // uniform per block -> scalar load

    int ih0; float wh0; lerp_coef(h0, HH, ih0, wh0);
    const int rtop = (ih0 + 3 < HH - 1) ? (ih0 + 3) : (HH - 1);
    const int nfl  = (rtop - ih0 + 1) * WH;          // 384 or 512 floats

    const float* rbase = readout + (((size_t)(b * CR + sel)) * HH + ih0) * WH;
    for (int e = t; e < nfl; e += 256)               // <=2 async b32 per lane
        stage_b32(rbase + e, &srow[e]);

    // Overlap: stream x (float4) while the async copy is in flight.
    const int r = t >> 6;               // row within tile, 0..3
    const int j = t & 63;               // float4 column, 0..63
    const int h = h0 + r;
    const size_t row4 = ((size_t)(b * C + c) * H + h) * (W / 4);
    const float4 xv = ((const float4*)x)[row4 + j];

    stage_wait();                       // s_wait_asynccnt 0 (this wave)
    __syncthreads();                    // all waves' rows visible

    int ih; float wh; lerp_coef(h, HH, ih, wh);
    const int lr = (ih - ih0) * WH;     // staged offset of row ih

    float rv[4];
#pragma unroll
    for (int k = 0; k < 4; ++k) {
        const int w = 4 * j + k;
        int iw; float ww; lerp_coef(w, WH, iw, ww);
        // H-interp first, then W (matches reference axis order)
        const float v0 = srow[lr + iw]     * (1.0f - wh) + srow[lr + WH + iw]     * wh;
        const float v1 = srow[lr + iw + 1] * (1.0f - wh) + srow[lr + WH + iw + 1] * wh;
        rv[k] = v0 * (1.0f - ww) + v1 * ww;
    }

    const float w0 = weight[2 * c];
    const float w1 = weight[2 * c + 1];
    const float bb = bias[c];

    float4 o;
    o.x = gate_blend(xv.x, rv[0], w0, w1, bb);
    o.y = gate_blend(xv.y, rv[1], w0, w1, bb);
    o.z = gate_blend(xv.z, rv[2], w0, w1, bb);
    o.w = gate_blend(xv.w, rv[3], w0, w1, bb);
    ((float4*)out)[row4 + j] = o;
}

extern "C" void kernel_launch(void* const* d_in, const int* in_sizes, int n_in,
                              void* d_out, int out_size, void* d_ws, size_t ws_size,
                              hipStream_t stream) {
    (void)in_sizes; (void)n_in; (void)out_size; (void)ws_size;

    const float* x       = (const float*)d_in[0];  // [8,64,256,256]
    const float* readout = (const float*)d_in[1];  // [8,128,128,128]
    const float* weight  = (const float*)d_in[2];  // [64,2,1,1]
    const float* bias    = (const float*)d_in[3];  // [64]
    float*       out     = (float*)d_out;          // [8,64,256,256]

    // scratch layout (all fully rewritten every call -> deterministic)
    float* cw     = (float*)d_ws;            // 128 floats
    float* scores = cw + HH;                 // B*CR = 1024 floats
    int*   idx    = (int*)(scores + B * CR); // B*C = 512 ints

    k_colsum<<<1, HH, 0, stream>>>(cw);
    k_scores<<<B * CR, 256, 0, stream>>>(readout, cw, scores);
    k_topk<<<B, CR, 0, stream>>>(scores, idx);
    k_fused<<<B * C * (H / 4), 256, 0, stream>>>(x, readout, weight, bias, idx, out);
}